// SpecialOrthogonal_34445637714305
// MI455X (gfx1250) — compile-verified
//
#include <hip/hip_runtime.h>

// SO(64) batched expm: K = skew(theta)/2^10, Taylor-12, 10 squarings.
// v5: single TRANSPOSED LDS layout. Every matrix is stored column-major
// (Mt[col*S + row]) so both WMMA operands load as naturally-contiguous
// register pairs: B-operand = ds_load_b64 of adjacent dwords, A-operand =
// ds_load_2addr_b32 pair one row-stride apart. No operand marshalling movs.
// Stride 68 makes the dominant b64 loads LDS-bank conflict-free.
// One workgroup (128 threads = 4 waves) per matrix; each wave computes a
// 16x64 strip (4 tiles) via V_WMMA_F32_16X16X4_F32, A-panel reused 4x.

typedef __attribute__((ext_vector_type(2))) float v2f;
typedef __attribute__((ext_vector_type(4))) float v4f;
typedef __attribute__((ext_vector_type(8))) float v8f;

#define SO_DIM 64
#define SO_NT 2016
#define LDS_S 68          // padded stride: b64 operand loads are bank-conflict-free
#define TAYLOR_ORDER 12
#define SQUARINGS 10

extern "C" __device__ float __builtin_amdgcn_rcpf(float);

// Strip matmul: D(16x64 strip ti) = X(64x64) * Y(64x64), X/Y stored
// TRANSPOSED in LDS (Mt[col*LDS_S + row]). m = lane%16, hi = lane/16.
// A operand (VGPR0=K, VGPR1=K+1, row M=m):  Xt[k*S + row], Xt[(k+1)*S + row]
// B operand (VGPR0=row K, VGPR1=row K+1, col N=m): Yt[n*S + k], Yt[n*S + k+1]
__device__ __forceinline__ void strip_mm64_t(const float* Xt, const float* Yt,
                                             int ti, int m, int hi, v8f d[4]) {
  const int row = ti * 16 + m;

  // A-panel: 16 operand pairs, each a 2addr_b32 (two dwords one stride apart).
  v2f a[16];
#pragma unroll
  for (int t = 0; t < 16; ++t) {
    const int ka = 4 * t + 2 * hi;
    v2f av;
    av.x = Xt[ka * LDS_S + row];
    av.y = Xt[ka * LDS_S + LDS_S + row];
    a[t] = av;
  }

#pragma unroll
  for (int tj = 0; tj < 4; ++tj) {
    const float* yk = Yt + (tj * 16 + m) * LDS_S + 2 * hi;
    v2f bv[16];
#pragma unroll
    for (int t = 0; t < 16; ++t)
      bv[t] = *(const v2f*)(yk + 4 * t);   // adjacent dwords -> ds_load_b64
    v8f acc = {};
#pragma unroll
    for (int t = 0; t < 16; ++t)
      acc = __builtin_amdgcn_wmma_f32_16x16x4_f32(
          false, a[t], false, bv[t], (short)0, acc, false, false);
    d[tj] = acc;
  }
}

// Store a 16x64 strip result transposed: per tile, 8 contiguous dwords/lane.
__device__ __forceinline__ void store_strip_t(float* Dt, int ti, int m, int hi,
                                              const v8f d[4]) {
  float* base = Dt + m * LDS_S + ti * 16 + 8 * hi;
#pragma unroll
  for (int tj = 0; tj < 4; ++tj) {
    v4f lo, hi4;
#pragma unroll
    for (int v = 0; v < 4; ++v) { lo[v] = d[tj][v]; hi4[v] = d[tj][v + 4]; }
    *(v4f*)(base + tj * 16 * LDS_S)     = lo;   // ds_store_b128
    *(v4f*)(base + tj * 16 * LDS_S + 4) = hi4;  // ds_store_b128
  }
}

__global__ __launch_bounds__(128)
void so_expm_kernel(const float* __restrict__ theta, float* __restrict__ out) {
  __shared__ float B0[SO_DIM * LDS_S];   // ping
  __shared__ float B1[SO_DIM * LDS_S];   // pong
  __shared__ float AT[SO_DIM * LDS_S];   // scaled A, transposed (persists Taylor)

  const int b    = blockIdx.x;
  const int tid  = threadIdx.x;
  const int lane = tid & 31;
  const int ti   = tid >> 5;    // wave = strip row 0..3
  const int m    = lane & 15;
  const int hi   = lane >> 4;

  const float scale = 1.0f / 1024.0f;  // 2^-10
  const float* th = theta + (size_t)b * SO_NT;

  // Build A^T (scaled antisymmetric K, column-major) in LDS. triu ordering:
  // idx(i,j) = i*64 - i*(i+1)/2 + (j-i-1); K[j,i]=+theta, K[i,j]=-theta (i<j).
  for (int e = tid; e < SO_DIM * SO_DIM; e += 128) {
    const int r = e >> 6;
    const int c = e & 63;
    float v = 0.0f;
    if (r < c) {
      v = -th[r * 64 - (r * (r + 1)) / 2 + (c - r - 1)] * scale;
    } else if (r > c) {
      v = th[c * 64 - (c * (c + 1)) / 2 + (r - c - 1)] * scale;
    }
    AT[c * LDS_S + r] = v;   // transposed store
  }
  __syncthreads();

  // R = I + A in registers (4 tiles/wave). A[row][col] = AT[col*S + row];
  // per lane the 8 rows are contiguous in AT -> vector loads.
  v8f R[4];
#pragma unroll
  for (int tj = 0; tj < 4; ++tj) {
    const float* p = AT + (tj * 16 + m) * LDS_S + ti * 16 + 8 * hi;
#pragma unroll
    for (int v = 0; v < 8; ++v) {
      const int row = ti * 16 + v + 8 * hi;
      const int col = tj * 16 + m;
      float x = p[v];
      if (row == col) x += 1.0f;
      R[tj][v] = x;
    }
  }

  // Taylor: T_1 = A; for k=2..12: T_k = (T_{k-1} @ A)/k, R += T_k.
  // X = curT (transposed), Y = AT always. Writes alternate B0,B1,B0,...
  const float* curT = AT;
  int wi = 0;                       // 0 -> write B0, 1 -> write B1
  for (int k = 2; k <= TAYLOR_ORDER; ++k) {
    v8f d[4];
    strip_mm64_t(curT, AT, ti, m, hi, d);
    const float inv = __builtin_amdgcn_rcpf((float)k);
#pragma unroll
    for (int tj = 0; tj < 4; ++tj)
#pragma unroll
      for (int v = 0; v < 8; ++v) d[tj][v] *= inv;
#pragma unroll
    for (int tj = 0; tj < 4; ++tj)
#pragma unroll
      for (int v = 0; v < 8; ++v) R[tj][v] += d[tj][v];
    float* dst = wi ? B1 : B0;
    store_strip_t(dst, ti, m, hi, d);
    __syncthreads();
    curT = dst;
    wi ^= 1;
  }
  // 11 iterations: writes B0,B1,...,B0 (k=12 writes B0, last read was B1).

  // Park R (transposed) in B1 — safe, all waves past the k=12 barrier.
  store_strip_t(B1, ti, m, hi, R);
  __syncthreads();

  // 9 squarings through LDS (read B1->write B0, then alternate), 10th to HBM.
  int rd = 1;
  for (int s = 0; s < SQUARINGS - 1; ++s) {
    const float* Rm = rd ? B1 : B0;
    v8f d[4];
    strip_mm64_t(Rm, Rm, ti, m, hi, d);
    float* dst = rd ? B0 : B1;
    store_strip_t(dst, ti, m, hi, d);
    __syncthreads();
    rd ^= 1;
  }
  {
    const float* Rm = rd ? B1 : B0;
    v8f d[4];
    strip_mm64_t(Rm, Rm, ti, m, hi, d);
    float* o = out + (size_t)b * SO_DIM * SO_DIM;
    const int orow0 = (ti * 16 + 8 * hi) * SO_DIM + m;
#pragma unroll
    for (int tj = 0; tj < 4; ++tj)
#pragma unroll
      for (int v = 0; v < 8; ++v)
        o[orow0 + v * SO_DIM + tj * 16] = d[tj][v];
  }
}

extern "C" void kernel_launch(void* const* d_in, const int* in_sizes, int n_in,
                              void* d_out, int out_size, void* d_ws, size_t ws_size,
                              hipStream_t stream) {
  (void)n_in; (void)d_ws; (void)ws_size; (void)out_size;
  const float* theta = (const float*)d_in[0];
  float* out = (float*)d_out;
  const int batch = in_sizes[0] / SO_NT;  // 8192
  so_expm_kernel<<<batch, 128, 0, stream>>>(theta, out);
}